// E_GAT_24438363914888
// MI455X (gfx1250) — compile-verified
//
#include <hip/hip_runtime.h>
#include <cstdint>
#include <cstddef>

// Problem constants (match reference)
#define BATCH 8
#define NN    2048
#define DD    128
#define NEG_SLOPE 0.2f

typedef float v2f __attribute__((ext_vector_type(2)));
typedef float v8f __attribute__((ext_vector_type(8)));

// D(16x16,f32) = A(16x4,f32) * B(4x16,f32) + C  — CDNA5 wave32 WMMA
__device__ __forceinline__ v8f wmma4_f32(v2f a, v2f b, v8f c) {
    return __builtin_amdgcn_wmma_f32_16x16x4_f32(
        /*neg_a=*/false, a, /*neg_b=*/false, b,
        /*c_mod=*/(short)0, c, /*reuse_a=*/false, /*reuse_b=*/false);
}

// ---------------------------------------------------------------------------
// Kernel 1: H = h @ W   (flattened rows = B*N, K = D = 128)
// Block: 256 thr = 8 waves; block owns 16 rows x 128 cols; wave w -> cols 16w..16w+15
// ---------------------------------------------------------------------------
__global__ void __launch_bounds__(256) k_gemm_HW(const float* __restrict__ h,
                                                 const float* __restrict__ W,
                                                 float* __restrict__ H) {
    __shared__ float hs[16][DD + 2];              // +2 pad: avoid 16-way bank conflicts
    const int row0 = blockIdx.x * 16;             // row in (B*N, D) view
    const int lane = threadIdx.x & 31;
    const int wave = threadIdx.x >> 5;

    for (int t = threadIdx.x; t < 16 * DD; t += 256) {
        const int r = t >> 7, c = t & (DD - 1);
        hs[r][c] = h[(size_t)(row0 + r) * DD + c];
    }
    __syncthreads();

    const int m  = lane & 15;
    const int kr = (lane < 16) ? 0 : 2;           // K slot for this half-wave
    const int cb = wave * 16;                     // output column base
    v8f acc = {};
#pragma unroll 8
    for (int kk = 0; kk < DD; kk += 4) {
        v2f A, Bv;
        A.x  = hs[m][kk + kr];
        A.y  = hs[m][kk + kr + 1];
        Bv.x = W[(size_t)(kk + kr)     * DD + cb + m];
        Bv.y = W[(size_t)(kk + kr + 1) * DD + cb + m];
        acc = wmma4_f32(A, Bv, acc);
    }
    const int mb = (lane < 16) ? 0 : 8;           // C/D layout: vgpr r -> M = r + mb
#pragma unroll
    for (int r = 0; r < 8; ++r)
        H[(size_t)(row0 + mb + r) * DD + cb + m] = acc[r];
}

// ---------------------------------------------------------------------------
// Kernel 2: s1[row] = H[row,:]·a[0:128],  s2[row] = H[row,:]·a[128:256]
// One wave per row, 8 waves / block.
// ---------------------------------------------------------------------------
__global__ void __launch_bounds__(256) k_scores(const float* __restrict__ H,
                                                const float* __restrict__ a,
                                                float* __restrict__ s1,
                                                float* __restrict__ s2) {
    const size_t row  = (size_t)blockIdx.x * 8 + (threadIdx.x >> 5);
    const int    lane = threadIdx.x & 31;
    const float* hr   = H + row * DD;
    float d1 = 0.f, d2 = 0.f;
#pragma unroll
    for (int c = lane; c < DD; c += 32) {
        const float v = hr[c];
        d1 += v * a[c];
        d2 += v * a[DD + c];
    }
    for (int off = 16; off > 0; off >>= 1) {
        d1 += __shfl_down(d1, off, 32);
        d2 += __shfl_down(d2, off, 32);
    }
    if (lane == 0) { s1[row] = d1; s2[row] = d2; }
}

// ---------------------------------------------------------------------------
// Kernel 3: per-row streaming softmax stats over rank-1 masked logits.
// m[row] = max_j l(i,j),  d[row] = sum_j exp(l - m), skipping masked j.
// One wave per row; mask streamed with prefetch.
// ---------------------------------------------------------------------------
__global__ void __launch_bounds__(256) k_stats(const float* __restrict__ s1,
                                               const float* __restrict__ s2,
                                               const uint8_t* __restrict__ mask,
                                               float* __restrict__ mrow,
                                               float* __restrict__ drow) {
    const size_t row  = (size_t)blockIdx.x * 8 + (threadIdx.x >> 5);
    const int    lane = threadIdx.x & 31;
    const int    b    = (int)(row >> 11);         // row / N
    const float  si   = s1[row];
    const float* s2b  = s2 + (size_t)b * NN;
    const uint8_t* mk = mask + row * (size_t)NN;

    float m = -__builtin_inff(), d = 0.f;
    for (int j = lane; j < NN; j += 32) {
        __builtin_prefetch(mk + j + 1024, 0, 0);  // global_prefetch_b8 ahead
        if (mk[j]) continue;
        float l = si + s2b[j];
        l = (l > 0.f) ? l : l * NEG_SLOPE;
        if (l > m) { d = d * __expf(m - l) + 1.0f; m = l; }
        else       { d += __expf(l - m); }
    }
    // cross-lane (m,d) merge, -inf safe
    for (int off = 16; off > 0; off >>= 1) {
        const float mo = __shfl_down(m, off, 32);
        const float dn = __shfl_down(d, off, 32);
        const float mx = fmaxf(m, mo);
        const float da = (m  > -1e37f) ? d  * __expf(m  - mx) : 0.f;
        const float db = (mo > -1e37f) ? dn * __expf(mo - mx) : 0.f;
        d = da + db;
        m = mx;
    }
    if (lane == 0) { mrow[row] = m; drow[row] = d; }
}

// ---------------------------------------------------------------------------
// Kernel 4: out = elu(H + (P @ H) / d), P regenerated on the fly per 16x16 tile.
// Block: 256 thr = 8 waves; owns (batch b, 16 i-rows); wave w -> cols 16w..16w+15.
// Loops j in steps of 16; P tile produced cooperatively into LDS, consumed as
// the WMMA A-fragment; B-fragment read directly from the L2-resident H.
// ---------------------------------------------------------------------------
__global__ void __launch_bounds__(256) k_attn(const float* __restrict__ H,
                                              const uint8_t* __restrict__ mask,
                                              const float* __restrict__ s1,
                                              const float* __restrict__ s2,
                                              const float* __restrict__ mrow,
                                              const float* __restrict__ drow,
                                              float* __restrict__ out) {
    __shared__ float P[16][18];                   // +2 pad
    const int b    = blockIdx.x >> 7;             // 128 i-tiles per batch
    const int i0   = (blockIdx.x & 127) * 16;
    const int lane = threadIdx.x & 31;
    const int wave = threadIdx.x >> 5;
    const int ti   = threadIdx.x >> 4;            // P producer coords: 16x16
    const int tj   = threadIdx.x & 15;

    const size_t rowbase = (size_t)b * NN + i0;
    const float  si = s1[rowbase + ti];
    const float  mi = mrow[rowbase + ti];
    const uint8_t* mk = mask + (rowbase + ti) * (size_t)NN;
    const float* s2b = s2 + (size_t)b * NN;
    const float* Hb  = H  + (size_t)b * NN * DD;

    const int m  = lane & 15;
    const int kr = (lane < 16) ? 0 : 2;
    const int cb = wave * 16;
    v8f acc = {};

    for (int j0 = 0; j0 < NN; j0 += 16) {
        // ---- produce P[i][j] = mask ? 0 : exp(leakyrelu(s1+s2) - m_i) ----
        float p = 0.f;
        if (!mk[j0 + tj]) {
            float l = si + s2b[j0 + tj];
            l = (l > 0.f) ? l : l * NEG_SLOPE;
            p = __expf(l - mi);
        }
        P[ti][tj] = p;
        __syncthreads();
        // ---- consume: acc(16x16) += P(16x16) @ H[j0:j0+16, cb:cb+16] ----
#pragma unroll
        for (int kk = 0; kk < 4; ++kk) {
            v2f A, Bv;
            A.x = P[m][4 * kk + kr];
            A.y = P[m][4 * kk + kr + 1];
            const float* hrow = Hb + (size_t)(j0 + 4 * kk + kr) * DD + cb + m;
            Bv.x = hrow[0];
            Bv.y = hrow[DD];
            acc = wmma4_f32(A, Bv, acc);
        }
        __syncthreads();
    }

    // ---- epilogue: out = elu(H + acc / d) ----
    const int mb = (lane < 16) ? 0 : 8;
#pragma unroll
    for (int r = 0; r < 8; ++r) {
        const size_t row = rowbase + mb + r;
        const float dinv = 1.0f / drow[row];
        const float x = H[row * DD + cb + m] + acc[r] * dinv;
        out[row * DD + cb + m] = (x > 0.f) ? x : (__expf(x) - 1.0f);
    }
}

// ---------------------------------------------------------------------------
extern "C" void kernel_launch(void* const* d_in, const int* in_sizes, int n_in,
                              void* d_out, int out_size, void* d_ws, size_t ws_size,
                              hipStream_t stream) {
    const float*   h    = (const float*)d_in[0];   // (8,2048,128) f32
    const float*   W    = (const float*)d_in[1];   // (128,128)    f32
    const float*   a    = (const float*)d_in[2];   // (256,1)      f32
    const uint8_t* mask = (const uint8_t*)d_in[3]; // (8,2048,2048) bool
    float* out = (float*)d_out;                    // (8,2048,128) f32

    // workspace layout
    float* H    = (float*)d_ws;                    // 8*2048*128
    const size_t Hsz = (size_t)BATCH * NN * DD;
    float* s1   = H    + Hsz;                      // 8*2048
    float* s2   = s1   + (size_t)BATCH * NN;
    float* mrow = s2   + (size_t)BATCH * NN;
    float* drow = mrow + (size_t)BATCH * NN;

    (void)in_sizes; (void)n_in; (void)out_size; (void)ws_size;

    k_gemm_HW<<<BATCH * NN / 16, 256, 0, stream>>>(h, W, H);
    k_scores <<<BATCH * NN / 8,  256, 0, stream>>>(H, a, s1, s2);
    k_stats  <<<BATCH * NN / 8,  256, 0, stream>>>(s1, s2, mask, mrow, drow);
    k_attn   <<<BATCH * NN / 16, 256, 0, stream>>>(H, mask, s1, s2, mrow, drow, out);
}